// DSBlock_88587995447581
// MI455X (gfx1250) — compile-verified
//
#include <hip/hip_runtime.h>
#include <hip/hip_bf16.h>
#include <math.h>

// MI455X (gfx1250) DSBlock. GEMMs on the WMMA pipe in exact fp32
// (V_WMMA_F32_16X16X4_F32): ~22 GFLOP vs ~150MB traffic at 23.3 TB/s ->
// bandwidth bound, so fp32 matrix ops keep reference numerics for free.
// Weight tiles are staged LDS-side by the Tensor Data Mover (one
// tensor_load_to_lds per block, issued by wave 0, s_wait_tensorcnt, barrier),
// then all waves read fragments via ds loads. TDM OOB-zeroing pads the
// 63-row PE-projection weights to K=64 so the K-loop has no bounds checks.

typedef float v2f __attribute__((ext_vector_type(2)));
typedef float v8f __attribute__((ext_vector_type(8)));
typedef unsigned int v4u __attribute__((ext_vector_type(4)));
typedef unsigned int v8u __attribute__((ext_vector_type(8)));

#define CEILDIV(a, b) (((a) + (b) - 1) / (b))

namespace {

constexpr int Bn = 4, Np = 2048, Kn = 16;
constexpr int DIM = 256, DIM_SA = 64, DIM_CONV = 192;
constexpr int Rr = 8, Mm = 8, MLP_H = 1024;
constexpr int NV = 512;                       // R^3 voxels
constexpr long XBS = (long)DIM * Np;          // batch stride of (B,256,N)
constexpr long CBS = (long)DIM_CONV * Np;     // batch stride of (B,192,N)

__device__ __forceinline__ float gelu_exact(float v) {
  return 0.5f * v * (1.0f + erff(v * 0.70710678118654752f));
}

// ---- Tensor Data Mover: 2D f32 tile (global -> LDS), D# per ISA 08 §8 ----
// td0/td1: tensor dims (OOB rows/cols read as zero -> free K padding)
// tl0/tl1: tile dims (tl0 = contiguous elements per row)
// str0:    row stride in elements
__device__ __forceinline__ void tdm_load_2d(const float* g, float* lds,
                                            unsigned td0, unsigned td1,
                                            unsigned tl0, unsigned tl1,
                                            unsigned long long str0) {
  unsigned long long ga = (unsigned long long)(size_t)g;
  unsigned ldsa = (unsigned)(unsigned long long)(size_t)lds;  // low 32b = LDS offset
  v4u g0;
  g0.x = 1u;                                   // count=1, user descriptor
  g0.y = ldsa;                                 // lds_addr
  g0.z = (unsigned)ga;                         // global_addr[31:0]
  g0.w = (unsigned)((ga >> 32) & 0x1FFFFFFull) | (2u << 30);  // addr[56:32]|type=2
  v8u g1;
  g1[0] = 2u << 16;                            // wg_mask=0, data_size=4B
  g1[1] = (td0 & 0xFFFFu) << 16;               // tensor_dim0[15:0] @ bits63:48
  g1[2] = (td0 >> 16) | ((td1 & 0xFFFFu) << 16);
  g1[3] = (td1 >> 16) | (tl0 << 16);           // tile_dim0 @ bits127:112
  g1[4] = tl1;                                 // tile_dim1; tile_dim2=0
  g1[5] = (unsigned)str0;                      // tensor_dim0_stride[31:0]
  g1[6] = (unsigned)(str0 >> 32);
  g1[7] = 0u;
  asm volatile("tensor_load_to_lds %0, %1" ::"s"(g0), "s"(g1) : "memory");
}

//========================= WMMA f32 GEMM kernels =============================
// A-frag (16x4): lane l: m=l%16, holds K = 2*(l/16)+{0,1} in {v0,v1}
// B-frag (4x16): lane l: n=l%16, holds K = 2*(l/16)+{0,1}
// C/D (16x16):  lane l: col=l%16, vgpr v: row = v + 8*(l/16)
// EPI bits: 1 = +bias, 2 = +addsrc (same layout as Y), 4 = exact GELU
// Kd must be a multiple of 16 (padded); Kw = real weight-K rows (<= Kd).

// Channel-major output: Y[b, o, n] = sum_k W'[o,k] * X[b, ., n]
template <bool IN_RMAJOR, bool W_T, int EPI>
__global__ void gemm_co(const float* __restrict__ X, const float* __restrict__ W,
                        const float* __restrict__ bias,
                        const float* __restrict__ addsrc, float* __restrict__ Y,
                        int Kd, int Kw, int Nn, int O, long xbs, long ybs) {
  extern __shared__ float wsm[];  // !W_T: [16][Kd] rows o; W_T: [Kd][16] rows k
  const int lane = threadIdx.x & 31;
  const int wave = threadIdx.x >> 5;  // 4 waves / block (wave32)
  const int b = blockIdx.z;
  const int n0 = (blockIdx.x * 4 + wave) * 16;
  const int o0 = blockIdx.y * 16;
  const int lm = lane & 15;
  const int lh = lane >> 4;
  if (wave == 0) {  // wave-uniform TDM issue: stage this block's weight tile
    if (W_T)
      tdm_load_2d(W + o0, wsm, O, Kw, 16, Kd, O);
    else
      tdm_load_2d(W + (long)o0 * Kd, wsm, Kd, O, Kd, 16, Kd);
    __builtin_amdgcn_s_wait_tensorcnt(0);
  }
  __syncthreads();
  const float* Xb = X + (long)b * xbs;
  const int nb = n0 + lm;
  v8f acc{};
#pragma unroll 4
  for (int k = 0; k < Kd; k += 4) {
    const int ka = k + 2 * lh;
    if (!IN_RMAJOR && (k & 15) == 0 && k + 16 < Kd)
      __builtin_prefetch(&Xb[(long)(k + 16) * Nn + nb], 0, 1);  // global_prefetch_b8
    v2f av;
    if (W_T) {
      av.x = wsm[ka * 16 + lm];
      av.y = wsm[ka * 16 + 16 + lm];
    } else {
      av = *(const v2f*)(wsm + lm * Kd + ka);  // ds_load_b64
    }
    v2f bv;
    if (IN_RMAJOR) {
      bv = *(const v2f*)(Xb + (long)nb * Kd + ka);  // global b64
    } else {
      bv.x = Xb[(long)ka * Nn + nb];
      bv.y = Xb[(long)(ka + 1) * Nn + nb];
    }
    acc = __builtin_amdgcn_wmma_f32_16x16x4_f32(false, av, false, bv, (short)0,
                                                acc, false, false);
  }
#pragma unroll
  for (int v = 0; v < 8; v++) {
    const int o = o0 + v + 8 * lh;
    const int n = n0 + lm;
    float r = acc[v];
    if (EPI & 1) r += bias[o];
    if (EPI & 4) r = gelu_exact(r);
    const long yi = (long)b * ybs + (long)o * Nn + n;
    if (EPI & 2) r += addsrc[yi];
    Y[yi] = r;
  }
}

// Row-major output: Y[b, t, j] = sum_k X[b, t, k] * W'[k, j]
template <bool IN_RMAJOR, bool W_T, int EPI>
__global__ void gemm_ro(const float* __restrict__ X, const float* __restrict__ W,
                        const float* __restrict__ bias, float* __restrict__ Y,
                        int Kd, int Nn, int O, long xbs, long ybs) {
  extern __shared__ float wsm[];  // !W_T: [Kd][16] rows k; W_T: [16][Kd] rows j
  const int lane = threadIdx.x & 31;
  const int wave = threadIdx.x >> 5;
  const int b = blockIdx.z;
  const int t0 = (blockIdx.x * 4 + wave) * 16;  // token rows
  const int j0 = blockIdx.y * 16;               // output features
  const int lm = lane & 15;
  const int lh = lane >> 4;
  if (wave == 0) {
    if (W_T)
      tdm_load_2d(W + (long)j0 * Kd, wsm, Kd, O, Kd, 16, Kd);
    else
      tdm_load_2d(W + j0, wsm, O, Kd, 16, Kd, O);
    __builtin_amdgcn_s_wait_tensorcnt(0);
  }
  __syncthreads();
  const float* Xb = X + (long)b * xbs;
  const int tm = t0 + lm;
  v8f acc{};
#pragma unroll 4
  for (int k = 0; k < Kd; k += 4) {
    const int ka = k + 2 * lh;
    v2f av;
    if (IN_RMAJOR) {
      av = *(const v2f*)(Xb + (long)tm * Kd + ka);  // global b64
    } else {
      av.x = Xb[(long)ka * Nn + tm];
      av.y = Xb[(long)(ka + 1) * Nn + tm];
    }
    v2f bv;
    if (W_T) {
      bv = *(const v2f*)(wsm + lm * Kd + ka);  // ds_load_b64
    } else {
      bv.x = wsm[ka * 16 + lm];
      bv.y = wsm[ka * 16 + 16 + lm];
    }
    acc = __builtin_amdgcn_wmma_f32_16x16x4_f32(false, av, false, bv, (short)0,
                                                acc, false, false);
  }
#pragma unroll
  for (int v = 0; v < 8; v++) {
    const int t = t0 + v + 8 * lh;
    const int j = j0 + lm;
    float r = acc[v];
    if (EPI & 1) r += bias[j];
    Y[(long)b * ybs + (long)t * O + j] = r;
  }
}

//=========================== elementwise / misc ==============================

__global__ void zero_kernel(float* __restrict__ p, long n) {
  long gid = (long)blockIdx.x * blockDim.x + threadIdx.x;
  if (gid < n) p[gid] = 0.0f;
}

__global__ void copy_kernel(const float* __restrict__ s, float* __restrict__ d, long n) {
  long gid = (long)blockIdx.x * blockDim.x + threadIdx.x;
  if (gid < n) d[gid] = s[gid];
}

// x1 = x + pos_embed(xyz); resid = x1; xbn = bn1(x1)
__global__ void posembed_bn1(const float* __restrict__ x, const float* __restrict__ xyz,
                             const float* __restrict__ pw, const float* __restrict__ pb,
                             const float* __restrict__ g, const float* __restrict__ be,
                             const float* __restrict__ rm, const float* __restrict__ rv,
                             float* __restrict__ resid, float* __restrict__ xbn) {
  long gid = (long)blockIdx.x * blockDim.x + threadIdx.x;
  if (gid >= (long)Bn * DIM * Np) return;
  const int n = gid & (Np - 1);
  const int o = (gid >> 11) & (DIM - 1);
  const int b = gid >> 19;
  float v = x[gid] + pb[o];
#pragma unroll
  for (int d = 0; d < 3; d++) v += pw[o * 3 + d] * xyz[((long)b * 3 + d) * Np + n];
  resid[gid] = v;
  const float s = g[o] * rsqrtf(rv[o] + 1e-5f);
  xbn[gid] = v * s + (be[o] - rm[o] * s);
}

__global__ void bn_kernel(const float* __restrict__ in, float* __restrict__ out,
                          const float* __restrict__ g, const float* __restrict__ be,
                          const float* __restrict__ rm, const float* __restrict__ rv,
                          int C, long in_bs, long out_bs) {
  long gid = (long)blockIdx.x * blockDim.x + threadIdx.x;
  if (gid >= (long)Bn * C * Np) return;
  const int n = gid % Np;
  const int c = (gid / Np) % C;
  const int b = gid / ((long)Np * C);
  const float s = g[c] * rsqrtf(rv[c] + 1e-5f);
  out[(long)b * out_bs + (long)c * Np + n] =
      in[(long)b * in_bs + (long)c * Np + n] * s + (be[c] - rm[c] * s);
}

// Ksum = pa_kernel[0:192,:] + pa_kernel[192:384,:]
__global__ void ksum_kernel(const float* __restrict__ pa, float* __restrict__ ks) {
  long gid = (long)blockIdx.x * blockDim.x + threadIdx.x;
  const long tot = (long)DIM_CONV * (Mm * DIM_CONV);
  if (gid < tot) ks[gid] = pa[gid] + pa[gid + tot];
}

// out[b,o,n] = res_conv[b,o,n] + sum_{k,m} score*point[idx] - sum_m (sum_k score)*center
__global__ void paconv_gather(const float* __restrict__ point,
                              const float* __restrict__ center,
                              const float* __restrict__ score, const int* __restrict__ idx,
                              const float* __restrict__ resconv, float* __restrict__ out) {
  __shared__ float ssc[Kn * Mm];
  __shared__ float ssum[Mm];
  __shared__ int sidx[Kn];
  const int n = blockIdx.x, b = blockIdx.y, t = threadIdx.x;
  const long row = (long)b * Np + n;
  if (t < Kn * Mm) ssc[t] = score[row * (Kn * Mm) + t];
  if (t < Kn) sidx[t] = idx[row * Kn + t];
  __syncthreads();
  if (t < Mm) {
    float s = 0.f;
    for (int k = 0; k < Kn; k++) s += ssc[k * Mm + t];
    ssum[t] = s;
  }
  __syncthreads();
  const int o = t;  // 0..191
  float acc = 0.f;
  for (int k = 0; k < Kn; k++) {
    const float* prow = point + ((long)b * Np + sidx[k]) * (Mm * DIM_CONV) + o;
#pragma unroll
    for (int m = 0; m < Mm; m++) acc += ssc[k * Mm + m] * prow[(long)m * DIM_CONV];
  }
  const float* crow = center + row * (Mm * DIM_CONV) + o;
#pragma unroll
  for (int m = 0; m < Mm; m++) acc -= ssum[m] * crow[(long)m * DIM_CONV];
  const long oi = (long)b * XBS + (long)o * Np + n;
  out[oi] = resconv[oi] + acc;
}

// per-batch mean over N and max ||xyz - mean||  -> stats[b*4] = {mx,my,mz,maxd}
__global__ void vox_stats(const float* __restrict__ xyz, float* __restrict__ stats) {
  __shared__ float red[256];
  __shared__ float mean[3];
  const int b = blockIdx.x, t = threadIdx.x;
  const float* xb = xyz + (long)b * 3 * Np;
  for (int c = 0; c < 3; c++) {
    float s = 0.f;
    for (int n = t; n < Np; n += 256) s += xb[(long)c * Np + n];
    red[t] = s;
    __syncthreads();
    for (int ofs = 128; ofs > 0; ofs >>= 1) {
      if (t < ofs) red[t] += red[t + ofs];
      __syncthreads();
    }
    if (t == 0) mean[c] = red[0] * (1.0f / Np);
    __syncthreads();
  }
  float mx = 0.f;
  for (int n = t; n < Np; n += 256) {
    const float dx = xb[n] - mean[0];
    const float dy = xb[Np + n] - mean[1];
    const float dz = xb[2 * Np + n] - mean[2];
    mx = fmaxf(mx, sqrtf(dx * dx + dy * dy + dz * dz));
  }
  red[t] = mx;
  __syncthreads();
  for (int ofs = 128; ofs > 0; ofs >>= 1) {
    if (t < ofs) red[t] = fmaxf(red[t], red[t + ofs]);
    __syncthreads();
  }
  if (t == 0) {
    stats[b * 4 + 0] = mean[0];
    stats[b * 4 + 1] = mean[1];
    stats[b * 4 + 2] = mean[2];
    stats[b * 4 + 3] = red[0];
  }
}

__global__ void nc_flat_kernel(const float* __restrict__ xyz, const float* __restrict__ stats,
                               float* __restrict__ nc, int* __restrict__ flat) {
  int gid = blockIdx.x * blockDim.x + threadIdx.x;
  if (gid >= Bn * Np) return;
  const int b = gid >> 11, n = gid & (Np - 1);
  const float* st = stats + b * 4;
  const float inv = 1.0f / (2.0f * st[3]);
  int vi[3];
#pragma unroll
  for (int d = 0; d < 3; d++) {
    const float c = (xyz[((long)b * 3 + d) * Np + n] - st[d]) * inv + 0.5f;
    const float v = fminf(fmaxf(c * (float)Rr, 0.0f), (float)(Rr - 1));
    nc[((long)b * 3 + d) * Np + n] = v;
    vi[d] = (int)rintf(v);  // round-half-even, matches jnp.round
  }
  flat[gid] = (vi[0] * Rr + vi[1]) * Rr + vi[2];
}

__global__ void scatter_count(const int* __restrict__ flat, float* __restrict__ cnt) {
  int gid = blockIdx.x * blockDim.x + threadIdx.x;
  if (gid >= Bn * Np) return;
  atomicAdd(&cnt[(gid >> 11) * NV + flat[gid]], 1.0f);
}

__global__ void scatter_feat(const float* __restrict__ feat, long f_bs,
                             const int* __restrict__ flat, float* __restrict__ sums, int C) {
  long gid = (long)blockIdx.x * blockDim.x + threadIdx.x;
  if (gid >= (long)Bn * C * Np) return;
  const int n = gid % Np;
  const int c = (gid / Np) % C;
  const int b = gid / ((long)Np * C);
  atomicAdd(&sums[((long)b * C + c) * NV + flat[b * Np + n]],
            feat[(long)b * f_bs + (long)c * Np + n]);
}

__global__ void vox_finalize(float* __restrict__ sums, const float* __restrict__ cnt, int C) {
  long gid = (long)blockIdx.x * blockDim.x + threadIdx.x;
  if (gid >= (long)Bn * C * NV) return;
  const int v = gid % NV;
  const int b = gid / ((long)NV * C);
  sums[gid] /= fmaxf(cnt[b * NV + v], 1.0f);
}

// trilinear gather: out[b,c,n] = sum_corners w * grid[b, c, voxel]
__global__ void devox_kernel(const float* __restrict__ grid, const float* __restrict__ nc,
                             float* __restrict__ out, int C, long g_bs, long g_cs,
                             long g_vs, long o_bs) {
  long gid = (long)blockIdx.x * blockDim.x + threadIdx.x;
  if (gid >= (long)Bn * C * Np) return;
  const int n = gid % Np;
  const int c = (gid / Np) % C;
  const int b = gid / ((long)Np * C);
  const float x = nc[((long)b * 3 + 0) * Np + n];
  const float y = nc[((long)b * 3 + 1) * Np + n];
  const float z = nc[((long)b * 3 + 2) * Np + n];
  const int lx = (int)floorf(x), ly = (int)floorf(y), lz = (int)floorf(z);
  const float fx = x - lx, fy = y - ly, fz = z - lz;
  const int hx = min(lx + 1, Rr - 1), hy = min(ly + 1, Rr - 1), hz = min(lz + 1, Rr - 1);
  const float* gb = grid + (long)b * g_bs + (long)c * g_cs;
  float acc = 0.f;
#pragma unroll
  for (int dx = 0; dx < 2; dx++)
#pragma unroll
    for (int dy = 0; dy < 2; dy++)
#pragma unroll
      for (int dz = 0; dz < 2; dz++) {
        const int ix = dx ? hx : lx, iy = dy ? hy : ly, iz = dz ? hz : lz;
        const float w = (dx ? fx : 1.f - fx) * (dy ? fy : 1.f - fy) * (dz ? fz : 1.f - fz);
        acc += w * gb[(long)((ix * Rr + iy) * Rr + iz) * g_vs];
      }
  out[(long)b * o_bs + (long)c * Np + n] = acc;
}

// embed_pe of the voxel grid (B,3,512) + pose const -> (B,512,64), lane 63 zero-padded
__global__ void hfemb_kernel(const float* __restrict__ g3, float* __restrict__ out) {
  int gid = blockIdx.x * blockDim.x + threadIdx.x;
  if (gid >= Bn * NV) return;
  const int b = gid >> 9, v = gid & (NV - 1);
  float g[3];
#pragma unroll
  for (int d = 0; d < 3; d++) g[d] = g3[((long)b * 3 + d) * NV + v];
  float* o = out + (long)gid * 64;
#pragma unroll
  for (int d = 0; d < 3; d++) o[d] = g[d];
  float fr = 1.0f;
  for (int f = 0; f < 10; f++) {
#pragma unroll
    for (int d = 0; d < 3; d++) {
      o[3 + 6 * f + d] = sinf(g[d] * fr);
      o[3 + 6 * f + 3 + d] = cosf(g[d] * fr) + 1.0f;  // + pose(zeros)
    }
    fr *= 2.0f;
  }
  o[63] = 0.0f;  // K padding
}

// embed_pe of xyz points (B,3,2048) -> (B,2048,64), lane 63 zero-padded
__global__ void xyzemb_kernel(const float* __restrict__ xyz, float* __restrict__ out) {
  int gid = blockIdx.x * blockDim.x + threadIdx.x;
  if (gid >= Bn * Np) return;
  const int b = gid >> 11, n = gid & (Np - 1);
  float g[3];
#pragma unroll
  for (int d = 0; d < 3; d++) g[d] = xyz[((long)b * 3 + d) * Np + n];
  float* o = out + (long)gid * 64;
#pragma unroll
  for (int d = 0; d < 3; d++) o[d] = g[d];
  float fr = 1.0f;
  for (int f = 0; f < 10; f++) {
#pragma unroll
    for (int d = 0; d < 3; d++) {
      o[3 + 6 * f + d] = sinf(g[d] * fr);
      o[3 + 6 * f + 3 + d] = cosf(g[d] * fr);
    }
    fr *= 2.0f;
  }
  o[63] = 0.0f;  // K padding
}

// pe = embed_pe(0) @ pv_pe_w + pv_pe_b  (constant over space) -> fold with posfwd bias
__global__ void pe_const_kernel(const float* __restrict__ pe_w, const float* __restrict__ pe_b,
                                const float* __restrict__ fwd_b, float* __restrict__ cbias) {
  const int o = threadIdx.x;
  if (o >= DIM_SA) return;
  float s = pe_b[o];
  for (int f = 0; f < 10; f++)
#pragma unroll
    for (int d = 0; d < 3; d++) s += pe_w[(3 + 6 * f + 3 + d) * DIM_SA + o];
  cbias[o] = s + fwd_b[o];
}

__global__ void ln_kernel(const float* __restrict__ in, float* __restrict__ out,
                          const float* __restrict__ g, const float* __restrict__ be) {
  int gid = blockIdx.x * blockDim.x + threadIdx.x;
  if (gid >= Bn * NV) return;
  const float* r = in + (long)gid * DIM_SA;
  float mu = 0.f;
  for (int c = 0; c < DIM_SA; c++) mu += r[c];
  mu *= (1.0f / DIM_SA);
  float var = 0.f;
  for (int c = 0; c < DIM_SA; c++) {
    const float d = r[c] - mu;
    var += d * d;
  }
  var *= (1.0f / DIM_SA);
  const float inv = rsqrtf(var + 1e-5f);
  float* o = out + (long)gid * DIM_SA;
  for (int c = 0; c < DIM_SA; c++) o[c] = (r[c] - mu) * inv * g[c] + be[c];
}

// streaming-softmax attention over 512 voxel tokens, hd=8; out = attn + res
__global__ void attn_kernel(const float* __restrict__ qkv, const float* __restrict__ res,
                            float* __restrict__ out) {
  int gid = blockIdx.x * blockDim.x + threadIdx.x;
  if (gid >= Bn * 8 * NV) return;
  const int qi = gid & (NV - 1);
  const int h = (gid >> 9) & 7;
  const int b = gid >> 12;
  const float* base = qkv + (long)b * NV * (3 * DIM_SA);
  float q[8];
  const float* qp = base + (long)qi * (3 * DIM_SA) + h * 8;
#pragma unroll
  for (int d = 0; d < 8; d++) q[d] = qp[d];
  float m = -1e30f, s = 0.f, o[8] = {};
  const float scale = 0.35355339059327373f;  // 8^-0.5
  for (int kk = 0; kk < NV; kk++) {
    const float* kp = base + (long)kk * (3 * DIM_SA) + DIM_SA + h * 8;
    float dot = 0.f;
#pragma unroll
    for (int d = 0; d < 8; d++) dot += q[d] * kp[d];
    dot *= scale;
    const float mn = fmaxf(m, dot);
    const float cf = __expf(m - mn);
    const float p = __expf(dot - mn);
    s = s * cf + p;
    const float* vp = base + (long)kk * (3 * DIM_SA) + 2 * DIM_SA + h * 8;
#pragma unroll
    for (int d = 0; d < 8; d++) o[d] = o[d] * cf + p * vp[d];
    m = mn;
  }
  const float inv = 1.0f / s;
  float* op = out + ((long)b * NV + qi) * DIM_SA + h * 8;
  const float* rp = res + ((long)b * NV + qi) * DIM_SA + h * 8;
#pragma unroll
  for (int d = 0; d < 8; d++) op[d] = o[d] * inv + rp[d];
}

}  // namespace

//=============================================================================

extern "C" void kernel_launch(void* const* d_in, const int* in_sizes, int n_in,
                              void* d_out, int out_size, void* d_ws, size_t ws_size,
                              hipStream_t stream) {
  // setup_inputs() dict order: x, xyz, idx, xyz_score, then params in insertion order
  const float* x       = (const float*)d_in[0];
  const float* xyz     = (const float*)d_in[1];
  const int*   idx     = (const int*)d_in[2];
  const float* score   = (const float*)d_in[3];
  const float* pos_w   = (const float*)d_in[4];
  const float* pos_b   = (const float*)d_in[5];
  const float* bn1_g   = (const float*)d_in[6];
  const float* bn1_b   = (const float*)d_in[7];
  const float* bn1_rm  = (const float*)d_in[8];
  const float* bn1_rv  = (const float*)d_in[9];
  const float* conv1_w = (const float*)d_in[10];
  const float* conv1_b = (const float*)d_in[11];
  const float* conv2_w = (const float*)d_in[12];
  const float* conv2_b = (const float*)d_in[13];
  const float* bnc_g   = (const float*)d_in[14];
  const float* bnc_b   = (const float*)d_in[15];
  const float* bnc_rm  = (const float*)d_in[16];
  const float* bnc_rv  = (const float*)d_in[17];
  const float* pa_ker  = (const float*)d_in[18];
  const float* f1_w    = (const float*)d_in[19];
  const float* f1_b    = (const float*)d_in[20];
  const float* f2_w    = (const float*)d_in[21];
  const float* f2_b    = (const float*)d_in[22];
  const float* pvf_w   = (const float*)d_in[23];
  const float* pvf_b   = (const float*)d_in[24];
  const float* pe_w    = (const float*)d_in[25];
  const float* pe_b    = (const float*)d_in[26];
  const float* ln_g    = (const float*)d_in[27];
  const float* ln_b    = (const float*)d_in[28];
  const float* up_w    = (const float*)d_in[29];
  const float* up_b    = (const float*)d_in[30];
  const float* fb_w    = (const float*)d_in[31];
  const float* fb_b    = (const float*)d_in[32];
  const float* bn2_g   = (const float*)d_in[33];
  const float* bn2_b   = (const float*)d_in[34];
  const float* bn2_rm  = (const float*)d_in[35];
  const float* bn2_rv  = (const float*)d_in[36];
  const float* fc1_w   = (const float*)d_in[37];
  const float* fc1_b   = (const float*)d_in[38];
  const float* fc2_w   = (const float*)d_in[39];
  const float* fc2_b   = (const float*)d_in[40];
  float* out = (float*)d_out;

  float* ws = (float*)d_ws;
  size_t off = 0;
  auto A = [&](size_t n) { float* p = ws + off; off += n; return p; };
  float* resid  = A(2097152);    // (B,256,N)
  float* xbn    = A(2097152);    // bn1 out; reused for bn2 out
  float* c1     = A(2097152);    // conv1 out: qkv ch0-63, res_conv ch64-255
  float* cbn    = A(1572864);    // bnc(conv part)
  float* ksum   = A(294912);     // K_lo + K_hi (192x1536)
  float* point  = A(12582912);   // (B,N,1536)
  float* center = A(12582912);   // (B,N,1536)
  float* xcat   = A(2097152);    // concat [conv(192) | sa_out(64)]
  float* x2     = A(2097152);    // after conv2 residual
  float* hidden = A(8388608);    // (B,1024,N)
  float* hfdev  = A(1572864);    // devox hf (B,192,N)
  float* cnt    = A(2048);       // voxel counts (cnt/hfsum/sasum contiguous: one zero pass)
  float* hfsum  = A(6144);       // hf grid sums
  float* sasum  = A(131072);     // sa grid sums
  float* nc     = A(24576);      // (B,3,N)
  int*   flat   = (int*)A(8192); // (B,N)
  float* stats  = A(16);         // per-batch mean3 + maxdist
  float* hfemb  = A(131072);     // (B,512,64) K-padded
  float* hfg192 = A(393216);     // (B,192,512)
  float* satok  = A(131072);     // (B,512,64) res_sa
  float* lntok  = A(131072);
  float* qkvtok = A(393216);     // (B,512,192)
  float* safin  = A(131072);     // attn out + res (flat-viewed as (B,64,512))
  float* xyzemb = A(524288);     // (B,2048,64) K-padded
  float* cbias  = A(64);         // pv_posfwd_b + pe-const

  const long PBS = (long)Np * (Mm * DIM_CONV);  // point/center batch stride
  auto SH = [](int kd) { return (size_t)kd * 16 * sizeof(float); };  // TDM tile bytes

  // 1) pos-embed + residual + bn1
  posembed_bn1<<<8192, 256, 0, stream>>>(x, xyz, pos_w, pos_b, bn1_g, bn1_b, bn1_rm,
                                         bn1_rv, resid, xbn);
  // 2) conv1 (WMMA)
  gemm_co<false, false, 1><<<dim3(32, 16, Bn), 128, SH(DIM), stream>>>(
      xbn, conv1_w, conv1_b, nullptr, c1, DIM, DIM, Np, DIM, XBS, XBS);
  // 3) bn on conv slice
  bn_kernel<<<6144, 256, 0, stream>>>(c1 + (long)DIM_SA * Np, cbn, bnc_g, bnc_b, bnc_rm,
                                      bnc_rv, DIM_CONV, XBS, CBS);
  // 4) fold pa_kernel halves (point = X @ (K_lo+K_hi))
  ksum_kernel<<<1152, 256, 0, stream>>>(pa_ker, ksum);
  // 5) point / center GEMMs (WMMA): 8192x192x1536 each
  gemm_ro<false, false, 0><<<dim3(32, 96, Bn), 128, SH(DIM_CONV), stream>>>(
      cbn, ksum, nullptr, point, DIM_CONV, Np, Mm * DIM_CONV, CBS, PBS);
  gemm_ro<false, false, 0><<<dim3(32, 96, Bn), 128, SH(DIM_CONV), stream>>>(
      cbn, pa_ker, nullptr, center, DIM_CONV, Np, Mm * DIM_CONV, CBS, PBS);
  // 6) paconv gather (+res_conv) -> xcat channels 0..191
  paconv_gather<<<dim3(Np, Bn), DIM_CONV, 0, stream>>>(point, center, score, idx,
                                                       c1 + (long)DIM_SA * Np, xcat);
  // 7) shared voxelization (nc/flat/counts identical for both branches)
  vox_stats<<<Bn, 256, 0, stream>>>(xyz, stats);
  nc_flat_kernel<<<32, 256, 0, stream>>>(xyz, stats, nc, flat);
  zero_kernel<<<CEILDIV(139264, 256), 256, 0, stream>>>(cnt, 139264);
  scatter_count<<<32, 256, 0, stream>>>(flat, cnt);
  scatter_feat<<<96, 256, 0, stream>>>(xyz, (long)3 * Np, flat, hfsum, 3);
  scatter_feat<<<2048, 256, 0, stream>>>(c1, XBS, flat, sasum, DIM_SA);
  vox_finalize<<<24, 256, 0, stream>>>(hfsum, cnt, 3);
  vox_finalize<<<512, 256, 0, stream>>>(sasum, cnt, DIM_SA);
  // 8) hf branch: embed -> f1 (WMMA, K padded 63->64) -> devox -> f2 (fused add)
  hfemb_kernel<<<8, 256, 0, stream>>>(hfsum, hfemb);
  gemm_co<true, true, 1><<<dim3(8, 12, Bn), 128, SH(64), stream>>>(
      hfemb, f1_w, f1_b, nullptr, hfg192, 64, 63, NV, DIM_CONV, (long)NV * 64,
      (long)DIM_CONV * NV);
  devox_kernel<<<6144, 256, 0, stream>>>(hfg192, nc, hfdev, DIM_CONV,
                                         (long)DIM_CONV * NV, NV, 1, CBS);
  gemm_co<false, true, 3><<<dim3(32, 12, Bn), 128, SH(DIM_CONV), stream>>>(
      hfdev, f2_w, f2_b, xcat, xcat, DIM_CONV, DIM_CONV, Np, DIM_CONV, CBS, XBS);
  // 9) sa branch: posfwd (+ folded pe bias, WMMA) -> LN -> up (WMMA) -> attn
  pe_const_kernel<<<1, 64, 0, stream>>>(pe_w, pe_b, pvf_b, cbias);
  gemm_ro<false, true, 1><<<dim3(8, 4, Bn), 128, SH(DIM_SA), stream>>>(
      sasum, pvf_w, cbias, satok, DIM_SA, NV, DIM_SA, (long)DIM_SA * NV,
      (long)NV * DIM_SA);
  ln_kernel<<<8, 256, 0, stream>>>(satok, lntok, ln_g, ln_b);
  gemm_ro<true, false, 1><<<dim3(8, 12, Bn), 128, SH(DIM_SA), stream>>>(
      lntok, up_w, up_b, qkvtok, DIM_SA, NV, 3 * DIM_SA, (long)NV * DIM_SA,
      (long)NV * 3 * DIM_SA);
  attn_kernel<<<64, 256, 0, stream>>>(qkvtok, satok, safin);
  // 10) devox sa (reference reshape flat-reinterprets (B,512,64) as (B,64,512))
  devox_kernel<<<2048, 256, 0, stream>>>(safin, nc, xcat + (long)DIM_CONV * Np, DIM_SA,
                                         (long)NV * DIM_SA, NV, 1, XBS);
  // 11) hf2: embed xyz (K padded) -> pv_fb (WMMA, fused add into sa_out channels)
  xyzemb_kernel<<<32, 256, 0, stream>>>(xyz, xyzemb);
  gemm_co<true, true, 3><<<dim3(32, 4, Bn), 128, SH(64), stream>>>(
      xyzemb, fb_w, fb_b, xcat + (long)DIM_CONV * Np, xcat + (long)DIM_CONV * Np, 64,
      63, Np, DIM_SA, (long)Np * 64, XBS);
  // 12) conv2 + residual (WMMA)
  gemm_co<false, false, 3><<<dim3(32, 16, Bn), 128, SH(DIM), stream>>>(
      xcat, conv2_w, conv2_b, resid, x2, DIM, DIM, Np, DIM, XBS, XBS);
  // 13) MLP: bn2 -> fc1+GELU (WMMA) -> fc2 + residual (WMMA, direct to d_out)
  bn_kernel<<<8192, 256, 0, stream>>>(x2, xbn, bn2_g, bn2_b, bn2_rm, bn2_rv, DIM, XBS,
                                      XBS);
  gemm_co<false, false, 5><<<dim3(32, 64, Bn), 128, SH(DIM), stream>>>(
      xbn, fc1_w, fc1_b, nullptr, hidden, DIM, DIM, Np, MLP_H, XBS, (long)MLP_H * Np);
  gemm_co<false, false, 3><<<dim3(32, 16, Bn), 128, SH(MLP_H), stream>>>(
      hidden, fc2_w, fc2_b, x2, out, MLP_H, MLP_H, Np, DIM, (long)MLP_H * Np, XBS);
  // 14) second tuple element: xyz passthrough
  copy_kernel<<<96, 256, 0, stream>>>(xyz, out + (long)Bn * XBS, (long)Bn * 3 * Np);
}